// CvxpyLayerSudokuLearnA_3032246911381
// MI455X (gfx1250) — compile-verified
//
#include <hip/hip_runtime.h>
#include <math.h>

// ---- problem constants (from reference) ----
#define M_EQ   324      // num equality constraints
#define NT     21       // ceil(324/16) tiles -> padded 336
#define MP     336
#define YD     729      // y dimension
#define NTILES 231      // NT*(NT+1)/2 lower-triangular 16x16 tiles
#define QINV   10.0f    // 1/Q_PEN
#define EPSR   1e-6f
#define NITERS 20
#define NBATCH 128
#define KC     16       // K-chunk width for LDS-staged SYRK
#define NCHUNK 46       // ceil(729/16) -> 736 padded K

typedef float v2f __attribute__((ext_vector_type(2)));
typedef float v8f __attribute__((ext_vector_type(8)));

typedef __attribute__((address_space(1))) int gint_t;   // global
typedef __attribute__((address_space(3))) int lint_t;   // LDS

__device__ __forceinline__ int tile_base(int I, int J) {
    return ((I * (I + 1)) / 2 + J) << 8; // * 256 floats per 16x16 tile
}

// Async DMA one 336x16 chunk of A (unmasked) into LDS.
// Row-clamped / col-clamped so every global address stays inside A; the
// junk lanes are neutralized later by the d-mask (cols) / pad cleanup (rows).
__device__ __forceinline__ void stage_chunk_async(const float* __restrict__ A,
                                                  float* AsmBuf, int kbase, int tid)
{
    for (int p = tid; p < MP * KC; p += 256) {          // exactly 21 full passes
        int r = p >> 4;
        int c = p & 15;
        int gr = r < M_EQ ? r : M_EQ - 1;
        int gc = kbase + c; if (gc > YD - 1) gc = YD - 1;
        const float* g = A + (size_t)gr * YD + gc;
        float* l = AsmBuf + p;
        __builtin_amdgcn_global_load_async_to_lds_b32(
            (gint_t*)(uintptr_t)g,
            (lint_t*)(uint32_t)(uintptr_t)l,
            0, 0);
    }
}

// One workgroup per batch element. J kept as blocked lower triangle in LDS
// (231 tiles * 1KB = 236.5KB). WMMA f32 16x16x4 for the masked SYRK (operands
// from an async-staged, double-buffered LDS chunk of A) and for the Cholesky
// trailing updates; wave-level substitution for the triangular solves.
__global__ __launch_bounds__(256)
void newton_qp_kernel(const float* __restrict__ X,
                      const float* __restrict__ A,
                      const float* __restrict__ Bvec,
                      float* __restrict__ Out)
{
    extern __shared__ float smem[];
    float* Jt   = smem;                  // 59136 floats
    float* yL   = Jt + NTILES * 256;     // 736 (pad zeros past 729)
    float* dA   = yL + 736;              // 736 (d * (1/Q); pad zeros)
    float* nuL  = dA + 736;              // 336
    float* xs   = nuL + 336;             // 336 : rhs r, then solution delta
    float* AsmL = xs + 336;              // 2 * 336*16 = 10752 (A chunk, dbl-buffered)
    int*   tmap = (int*)(AsmL + 2 * MP * KC); // 232: t -> (I<<8)|J triangular LUT

    const int tid  = threadIdx.x;
    const int lane = tid & 31;
    const int wv   = tid >> 5;           // wave id 0..7 (wave32)
    const int b    = blockIdx.x;
    const float* xf = X + (size_t)b * YD;

    for (int e = tid; e < MP; e += 256) nuL[e] = 0.0f;
    for (int i = YD + tid; i < 736; i += 256) { yL[i] = 0.0f; dA[i] = 0.0f; }
    for (int t = tid; t < NTILES; t += 256) {           // one-time triangular LUT
        int I = 0, base = 0;
        while (base + I + 1 <= t) { base += I + 1; ++I; }
        tmap[t] = (I << 8) | (t - base);
    }
    __syncthreads();

    for (int it = 0; it <= NITERS; ++it) {
        // ---------- z = (xf - nu A)/Q ; y = relu(z) ; d/Q mask ----------
        for (int i = tid; i < YD; i += 256) {
            float acc = xf[i];
            const float* Ac = A + i;                    // coalesced across lanes
            for (int e = 0; e < M_EQ; ++e) acc -= nuL[e] * Ac[(size_t)e * YD];
            float z = acc * QINV;
            float y = z > 0.0f ? z : 0.0f;
            if (it == NITERS) Out[(size_t)b * YD + i] = y;
            yL[i] = y;
            dA[i] = z > 0.0f ? QINV : 0.0f;             // fold 1/Q into the SYRK mask
        }
        if (it == NITERS) break;
        __syncthreads();

        // ---------- r = y A^T - b (wave-per-row dot + shuffle reduce) ----------
        for (int e = wv; e < M_EQ; e += 8) {
            const float* Ar = A + (size_t)e * YD;
            float p = 0.0f;
            for (int i = lane; i < YD; i += 32) p += yL[i] * Ar[i];
            for (int off = 16; off > 0; off >>= 1) p += __shfl_xor(p, off, 32);
            if (lane == 0) xs[e] = p - Bvec[e];
        }
        for (int e = M_EQ + tid; e < MP; e += 256) xs[e] = 0.0f;
        __syncthreads();

        // ---------- J = A diag(d/Q) A^T : K-chunked LDS-staged SYRK ----------
        stage_chunk_async(A, AsmL, 0, tid);             // chunk 0
        __builtin_amdgcn_s_wait_asynccnt(0);
        __syncthreads();
        for (int c = 0; c < NCHUNK; ++c) {
            const int kbase = c * KC;
            const int cur = c & 1;
            if (c + 1 < NCHUNK)                         // overlap DMA of next chunk
                stage_chunk_async(A, AsmL + ((cur ^ 1) * MP * KC), kbase + KC, tid);

            const int Mi = lane & 15, hi = lane >> 4;
            v2f dk[4];
            #pragma unroll
            for (int s = 0; s < 4; ++s) {               // per-chunk mask pairs, hoisted
                int kk = kbase + 4 * s + (hi << 1);
                dk[s].x = dA[kk]; dk[s].y = dA[kk + 1];
            }
            const float* Ac0 = AsmL + cur * MP * KC;

            for (int t = wv; t < NTILES; t += 8) {
                int ij = tmap[t];
                int I = ij >> 8, Jn = ij & 255;
                const float* aE = Ac0 + (I * 16 + Mi) * KC;
                const float* aF = Ac0 + (Jn * 16 + Mi) * KC;
                float* tp = Jt + tile_base(I, Jn);
                v8f acc;
                if (c == 0) {
                    #pragma unroll
                    for (int v = 0; v < 8; ++v) acc[v] = 0.0f;
                } else {
                    #pragma unroll
                    for (int v = 0; v < 8; ++v) acc[v] = tp[(v + hi * 8) * 16 + Mi];
                }
                #pragma unroll
                for (int s = 0; s < 4; ++s) {
                    int ks = 4 * s + (hi << 1);
                    v2f av, bv;
                    av.x = aE[ks]     * dk[s].x;        // mask applied to A operand only
                    av.y = aE[ks + 1] * dk[s].y;
                    bv.x = aF[ks];
                    bv.y = aF[ks + 1];
                    acc = __builtin_amdgcn_wmma_f32_16x16x4_f32(
                            false, av, false, bv, (short)0, acc, false, false);
                }
                #pragma unroll
                for (int v = 0; v < 8; ++v) tp[(v + hi * 8) * 16 + Mi] = acc[v];
            }
            __builtin_amdgcn_s_wait_asynccnt(0);
            __syncthreads();
        }

        // ---------- pad cleanup + Tikhonov ----------
        for (int p = tid; p < 12 * NT * 16; p += 256) {        // zero pad rows 324..335
            int r = 4 + p / (NT * 16);
            int rem = p % (NT * 16);
            Jt[tile_base(NT - 1, rem >> 4) + r * 16 + (rem & 15)] = 0.0f;
        }
        __syncthreads();
        for (int p = tid; p < 16 * 12; p += 256) {             // pad cols of tile(20,20), unit diag
            int r = p / 12, c = 4 + p % 12;
            Jt[tile_base(NT - 1, NT - 1) + r * 16 + c] = (r == c) ? 1.0f : 0.0f;
        }
        __syncthreads();
        for (int e = tid; e < M_EQ; e += 256) {                // + eps I
            int Ti = e >> 4, c = e & 15;
            Jt[tile_base(Ti, Ti) + c * 17] += EPSR;
        }
        __syncthreads();

        // ---------- blocked Cholesky of J (in LDS) ----------
        for (int k = 0; k < NT; ++k) {
            if (tid < 16) {                             // diagonal factor, lane r owns row r
                float* T = Jt + tile_base(k, k);
                int r = tid;
                for (int j = 0; j < 16; ++j) {
                    float dv = sqrtf(fmaxf(T[j * 16 + j], 1e-30f));
                    float lrj = 0.0f;
                    if (r >= j) {
                        lrj = (r == j) ? dv : T[r * 16 + j] / dv;
                        T[r * 16 + j] = lrj;
                    }
                    for (int c = j + 1; c <= r; ++c)
                        T[r * 16 + c] -= lrj * T[c * 16 + j];
                }
            }
            __syncthreads();
            {                                           // panel: one thread per row
                int nrows = (NT - 1 - k) * 16;
                const float* Td = Jt + tile_base(k, k);
                for (int p = tid; p < nrows; p += 256) {
                    int i = k + 1 + (p >> 4);
                    float* P = Jt + tile_base(i, k) + (p & 15) * 16;
                    float w[16];
                    #pragma unroll
                    for (int c = 0; c < 16; ++c) w[c] = P[c];
                    #pragma unroll
                    for (int c = 0; c < 16; ++c) {
                        float s = w[c];
                        for (int t2 = 0; t2 < c; ++t2) s -= Td[c * 16 + t2] * w[t2];
                        w[c] = s / Td[c * 16 + c];
                    }
                    #pragma unroll
                    for (int c = 0; c < 16; ++c) P[c] = w[c];
                }
            }
            __syncthreads();
            {                                           // trailing: WMMA on LDS tiles
                int cnt = NT - 1 - k; cnt = cnt * (cnt + 1) / 2;
                int Mi = lane & 15, hi = lane >> 4;
                for (int p = wv; p < cnt; p += 8) {
                    int ij = tmap[p];                   // shifted-triangle prefix of same LUT
                    int i = k + 1 + (ij >> 8);
                    int j = k + 1 + (ij & 255);
                    const float* Li = Jt + tile_base(i, k);
                    const float* Lj = Jt + tile_base(j, k);
                    float* C = Jt + tile_base(i, j);
                    v8f acc;
                    #pragma unroll
                    for (int v = 0; v < 8; ++v) acc[v] = C[(v + hi * 8) * 16 + Mi];
                    #pragma unroll
                    for (int kc = 0; kc < 16; kc += 4) {
                        int kk = kc + (hi << 1);
                        v2f av, bv;
                        av.x = -Li[Mi * 16 + kk];       // negate A => C - L L^T
                        av.y = -Li[Mi * 16 + kk + 1];
                        bv.x =  Lj[Mi * 16 + kk];       // B[k][n] = Ljk[n][k]
                        bv.y =  Lj[Mi * 16 + kk + 1];
                        acc = __builtin_amdgcn_wmma_f32_16x16x4_f32(
                                false, av, false, bv, (short)0, acc, false, false);
                    }
                    #pragma unroll
                    for (int v = 0; v < 8; ++v) C[(v + hi * 8) * 16 + Mi] = acc[v];
                }
            }
            __syncthreads();
        }

        // ---------- solve L L^T delta = r (wave0 lanes 0..15, blocked subst) ----------
        if (tid < 16) {
            int r = tid;
            for (int I = 0; I < NT; ++I) {              // forward
                float accv = xs[I * 16 + r];
                for (int Jn = 0; Jn < I; ++Jn) {
                    const float* L2 = Jt + tile_base(I, Jn) + r * 16;
                    const float* xp = xs + Jn * 16;
                    #pragma unroll
                    for (int c = 0; c < 16; ++c) accv -= L2[c] * xp[c];
                }
                const float* Td = Jt + tile_base(I, I);
                xs[I * 16 + r] = accv;
                for (int j = 0; j < 16; ++j) {
                    float xj = xs[I * 16 + j] / Td[j * 16 + j];
                    if (r == j)      xs[I * 16 + j]  = xj;
                    else if (r > j)  xs[I * 16 + r] -= Td[r * 16 + j] * xj;
                }
            }
            for (int I = NT - 1; I >= 0; --I) {         // backward (L^T)
                float accv = xs[I * 16 + r];
                for (int Jn = I + 1; Jn < NT; ++Jn) {
                    const float* L2 = Jt + tile_base(Jn, I);
                    const float* xp = xs + Jn * 16;
                    #pragma unroll
                    for (int c = 0; c < 16; ++c) accv -= L2[c * 16 + r] * xp[c];
                }
                const float* Td = Jt + tile_base(I, I);
                xs[I * 16 + r] = accv;
                for (int j = 15; j >= 0; --j) {
                    float xj = xs[I * 16 + j] / Td[j * 16 + j];
                    if (r == j)      xs[I * 16 + j]  = xj;
                    else if (r < j)  xs[I * 16 + r] -= Td[j * 16 + r] * xj;
                }
            }
        }
        __syncthreads();

        // ---------- nu += delta ----------
        for (int e = tid; e < M_EQ; e += 256) nuL[e] += xs[e];
        __syncthreads();
    }
}

extern "C" void kernel_launch(void* const* d_in, const int* in_sizes, int n_in,
                              void* d_out, int out_size, void* d_ws, size_t ws_size,
                              hipStream_t stream) {
    (void)in_sizes; (void)n_in; (void)out_size; (void)d_ws; (void)ws_size;
    const float* X  = (const float*)d_in[0];   // (128,9,9,9)
    const float* A  = (const float*)d_in[1];   // (324,729)
    const float* Bv = (const float*)d_in[2];   // (324,)
    float* Out = (float*)d_out;                // (128,9,9,9)

    const size_t smem_bytes =
        (size_t)(NTILES * 256 + 736 + 736 + 336 + 336 + 2 * MP * KC + 232) * sizeof(float); // ~289 KB

    newton_qp_kernel<<<NBATCH, 256, smem_bytes, stream>>>(X, A, Bv, Out);
}